// HierarchicalLoss_23862838297190
// MI455X (gfx1250) — compile-verified
//
#include <hip/hip_runtime.h>

typedef float v2f __attribute__((ext_vector_type(2)));
typedef float v8f __attribute__((ext_vector_type(8)));

// ---------------------------------------------------------------------------
// 1) Global average pool: one wave32 per (b,c) row of HW contiguous floats.
//    Coalesced b32 lane loads, wave shuffle reduction. Bandwidth-bound stage.
// ---------------------------------------------------------------------------
__global__ void pool_kernel(const float* __restrict__ feat,
                            float* __restrict__ pooled,
                            int HW, float inv, long nrows)
{
    long gtid = (long)blockIdx.x * blockDim.x + threadIdx.x;
    long row  = gtid >> 5;
    int  lane = threadIdx.x & 31;
    if (row >= nrows) return;

    const float* p = feat + row * (long)HW;
    float s = 0.0f;
    for (int i = lane; i < HW; i += 32)
        s += p[i];
    #pragma unroll
    for (int o = 16; o > 0; o >>= 1)
        s += __shfl_xor(s, o, 32);
    if (lane == 0)
        pooled[row] = s * inv;
}

// ---------------------------------------------------------------------------
// 2) Linear head via V_WMMA_F32_16X16X4_F32: one wave per 16x16 output tile.
//    logits[m][n] = sum_k pooled[m][k] * W[n][k] + bias[n]
//    A-frag (16x4 f32): lane<16 -> M=lane, K={k,k+1}; lane>=16 -> K={k+2,k+3}
//    B-frag (4x16 f32): N=lane%16, lane<16 -> K={k,k+1}; lane>=16 -> K={k+2,k+3}
//    D (16x16 f32): vgpr i, lane L -> M = i + 8*(L/16), N = L%16
// ---------------------------------------------------------------------------
__global__ void wmma_head_kernel(const float* __restrict__ pooled, // [B, C]
                                 const float* __restrict__ W,      // [Kc, C]
                                 const float* __restrict__ bias,   // [Kc]
                                 float* __restrict__ logits,       // [B, Npad]
                                 int C, int Kc, int Npad)
{
    const int mtile = blockIdx.x;
    const int ntile = blockIdx.y;
    const int lane  = threadIdx.x;      // 0..31, block is exactly one wave
    const int half  = lane >> 4;        // 0 or 1
    const int lmod  = lane & 15;

    const int mbase = mtile * 16;
    const int nbase = ntile * 16;
    const int n     = nbase + lmod;
    const bool nvalid = (n < Kc);
    const int ncl   = nvalid ? n : 0;

    const float* arow = pooled + (size_t)(mbase + lmod) * C + half * 2;
    const float* brow = W      + (size_t)ncl * C            + half * 2;

    v8f acc = {};
    #pragma unroll 4
    for (int k = 0; k < C; k += 4) {
        v2f a, b;
        a.x = arow[k];
        a.y = arow[k + 1];
        b.x = brow[k];
        b.y = brow[k + 1];
        if (!nvalid) { b.x = 0.0f; b.y = 0.0f; }
        // D = A*B + C   (emits v_wmma_f32_16x16x4_f32)
        acc = __builtin_amdgcn_wmma_f32_16x16x4_f32(
            /*neg_a=*/false, a, /*neg_b=*/false, b,
            /*c_mod=*/(short)0, acc, /*reuse_a=*/false, /*reuse_b=*/false);
    }

    if (nvalid) {
        const float bv = bias[n];
        #pragma unroll
        for (int i = 0; i < 8; ++i) {
            const int m = mbase + i + 8 * half;
            logits[(size_t)m * Npad + n] = acc[i] + bv;
        }
    }
}

// ---------------------------------------------------------------------------
// 3) Weighted CE partials: one wave32 per sample; max + logsumexp over Kc
//    logits, then per-sample (w*nll, w) written out (deterministic).
// ---------------------------------------------------------------------------
__global__ void ce_kernel(const float* __restrict__ logits, // [B, Npad]
                          int Npad, int Kc,
                          const int* __restrict__ target,
                          const int* __restrict__ lut,
                          const float* __restrict__ cw,
                          float* __restrict__ nll_out,      // [B]
                          float* __restrict__ w_out,        // [B]
                          int B)
{
    int gtid = blockIdx.x * blockDim.x + threadIdx.x;
    int s    = gtid >> 5;
    int lane = threadIdx.x & 31;
    if (s >= B) return;

    const float* row = logits + (size_t)s * Npad;

    float mx = -__builtin_inff();
    for (int k = lane; k < Kc; k += 32)
        mx = fmaxf(mx, row[k]);
    #pragma unroll
    for (int o = 16; o > 0; o >>= 1)
        mx = fmaxf(mx, __shfl_xor(mx, o, 32));

    float se = 0.0f;
    for (int k = lane; k < Kc; k += 32)
        se += expf(row[k] - mx);
    #pragma unroll
    for (int o = 16; o > 0; o >>= 1)
        se += __shfl_xor(se, o, 32);

    if (lane == 0) {
        const int t = lut[target[s]];
        const float nll = mx + logf(se) - row[t];
        const float w = cw[t];
        nll_out[s] = w * nll;
        w_out[s]   = w;
    }
}

// ---------------------------------------------------------------------------
// 4) Deterministic final reduction of the 4 length-B partial arrays.
// ---------------------------------------------------------------------------
__global__ void finalize_kernel(const float* __restrict__ nll0,
                                const float* __restrict__ w0,
                                const float* __restrict__ nll1,
                                const float* __restrict__ w1,
                                float* __restrict__ out, int B)
{
    __shared__ float sm[4 * 256];
    const int t = threadIdx.x;
    float a = 0.f, b = 0.f, c = 0.f, d = 0.f;
    for (int i = t; i < B; i += 256) {
        a += nll0[i]; b += w0[i]; c += nll1[i]; d += w1[i];
    }
    sm[t] = a; sm[256 + t] = b; sm[512 + t] = c; sm[768 + t] = d;
    __syncthreads();
    for (int o = 128; o > 0; o >>= 1) {
        if (t < o) {
            sm[t]       += sm[t + o];
            sm[256 + t] += sm[256 + t + o];
            sm[512 + t] += sm[512 + t + o];
            sm[768 + t] += sm[768 + t + o];
        }
        __syncthreads();
    }
    if (t == 0)
        out[0] = sm[0] / sm[256] + sm[512] / sm[768];
}

// ---------------------------------------------------------------------------
// Launch
// ---------------------------------------------------------------------------
extern "C" void kernel_launch(void* const* d_in, const int* in_sizes, int n_in,
                              void* d_out, int out_size, void* d_ws, size_t ws_size,
                              hipStream_t stream)
{
    (void)in_sizes; (void)n_in; (void)out_size; (void)ws_size;

    constexpr int B    = 1024;
    constexpr int C0   = 512,  H0 = 14, HW0 = H0 * H0;   // 196
    constexpr int C1   = 1024, H1 = 7,  HW1 = H1 * H1;   // 49
    constexpr int K0   = 100,  NP0 = 112;                // pad to 7 N-tiles
    constexpr int K1   = 50,   NP1 = 64;                 // pad to 4 N-tiles

    const float* feat0 = (const float*)d_in[0];
    const float* feat1 = (const float*)d_in[1];
    const float* W0    = (const float*)d_in[2];
    const float* b0    = (const float*)d_in[3];
    const float* W1    = (const float*)d_in[4];
    const float* b1    = (const float*)d_in[5];
    const int*   lut0  = (const int*)d_in[6];
    const int*   lut1  = (const int*)d_in[7];
    const float* cw0   = (const float*)d_in[8];
    const float* cw1   = (const float*)d_in[9];
    const int*   tgt   = (const int*)d_in[10];

    // workspace layout (floats)
    float* ws       = (float*)d_ws;
    float* pooled0  = ws;                               // B*C0
    float* pooled1  = pooled0 + (size_t)B * C0;         // B*C1
    float* logits0  = pooled1 + (size_t)B * C1;         // B*NP0
    float* logits1  = logits0 + (size_t)B * NP0;        // B*NP1
    float* nll0     = logits1 + (size_t)B * NP1;        // B
    float* wgt0     = nll0 + B;                         // B
    float* nll1     = wgt0 + B;                         // B
    float* wgt1     = nll1 + B;                         // B

    // 1) pooling (bandwidth-dominant: ~616 MB streamed once)
    {
        long rows0 = (long)B * C0;                      // 524288 waves
        long rows1 = (long)B * C1;                      // 1048576 waves
        int blocks0 = (int)((rows0 * 32 + 255) / 256);
        int blocks1 = (int)((rows1 * 32 + 255) / 256);
        pool_kernel<<<blocks0, 256, 0, stream>>>(feat0, pooled0, HW0, 1.0f / HW0, rows0);
        pool_kernel<<<blocks1, 256, 0, stream>>>(feat1, pooled1, HW1, 1.0f / HW1, rows1);
    }

    // 2) WMMA linear heads (fp32 16x16x4), one wave per 16x16 tile
    wmma_head_kernel<<<dim3(B / 16, NP0 / 16), 32, 0, stream>>>(
        pooled0, W0, b0, logits0, C0, K0, NP0);
    wmma_head_kernel<<<dim3(B / 16, NP1 / 16), 32, 0, stream>>>(
        pooled1, W1, b1, logits1, C1, K1, NP1);

    // 3) CE partials: one wave per sample
    {
        int blocks = (B * 32) / 256;                    // 128
        ce_kernel<<<blocks, 256, 0, stream>>>(logits0, NP0, K0, tgt, lut0, cw0, nll0, wgt0, B);
        ce_kernel<<<blocks, 256, 0, stream>>>(logits1, NP1, K1, tgt, lut1, cw1, nll1, wgt1, B);
    }

    // 4) deterministic scalar combine
    finalize_kernel<<<1, 256, 0, stream>>>(nll0, wgt0, nll1, wgt1, (float*)d_out, B);
}